// X1SelfAttention_35227321762416
// MI455X (gfx1250) — compile-verified
//
#include <hip/hip_runtime.h>

typedef __attribute__((ext_vector_type(2))) float v2f;
typedef __attribute__((ext_vector_type(8))) float v8f;
typedef __attribute__((ext_vector_type(4))) unsigned int u32x4;
typedef __attribute__((ext_vector_type(8))) int i32x8;
typedef __attribute__((ext_vector_type(4))) int i32x4;

#define B_  32
#define S_  2048
#define H_  1024
#define A_  512
#define X0_ 1024

#define ROWS 64                 // s-rows per workgroup
#define KC   128                // K-chunk staged in LDS
#define LDS_STRIDE (KC + 4)     // 132 floats = 128 data + 4 pad (TDM pad reproduces this)

// workspace layout (floats)
#define WS_WHT 0
#define WS_A0  (A_ * H_)               // 524288
#define WS_SC  (WS_A0 + B_ * A_)       // 540672  (total ~2.42 MB)

// ---------------------------------------------------------------------------
// Tensor Data Mover: DMA a ROWS x KC fp32 tile (row stride H_) into LDS with
// hardware padding of 4 DWORDs after every 128 DWORDs (-> 132-float LDS rows).
// D# layout per CDNA5 ISA ch.8 (groups 0/1; groups 2/3 unused for 2D).
// This toolchain's builtin is the 6-arg clang-23 form.
__device__ __forceinline__ void tdm_load_tile(const float* gsrc, unsigned lds_addr) {
    unsigned long long ga = (unsigned long long)(uintptr_t)gsrc;
    u32x4 g0;
    g0[0] = 1u;                                        // count=1, user descriptor
    g0[1] = lds_addr;                                  // LDS byte address
    g0[2] = (unsigned)(ga & 0xFFFFFFFFu);              // global_addr[31:0]
    g0[3] = (unsigned)((ga >> 32) & 0x1FFFFFFu)        // global_addr[56:32]
          | (2u << 30);                                // type=2 ("image")
    i32x8 g1;
    g1[0] = (2 << 16)                                  // data_size = 4 bytes
          | (1 << 20)                                  // pad_enable
          | (6 << 22)                                  // pad_interval: 128 DWORDs
          | (3 << 25);                                 // pad_amount: 4 DWORDs
    g1[1] = (int)((H_ & 0xFFFF) << 16);                // tensor_dim0[15:0] (=1024)
    g1[2] = (int)(((unsigned)H_ >> 16) | ((S_ & 0xFFFF) << 16)); // dim0 hi / dim1 lo
    g1[3] = (int)(((unsigned)S_ >> 16) | (KC << 16));  // dim1 hi / tile_dim0 = 128
    g1[4] = ROWS;                                      // tile_dim1 = 64, tile_dim2 = 0
    g1[5] = H_;                                        // tensor_dim0_stride[31:0] = 1024
    g1[6] = 0;                                         // stride hi / dim1_stride lo
    g1[7] = 0;
    i32x4 z4 = {0, 0, 0, 0};
    i32x8 z8 = {0, 0, 0, 0, 0, 0, 0, 0};
    __builtin_amdgcn_tensor_load_to_lds(g0, g1, z4, z4, z8, 0);
}

// ---------------------------------------------------------------------------
// Wh [A,H] -> WhT [H,A] so B-fragment loads are coalesced across lanes (N dim)
__global__ __launch_bounds__(256) void k_transpose(const float* __restrict__ Wh,
                                                   float* __restrict__ WhT) {
    int idx = blockIdx.x * 256 + threadIdx.x;   // over A_*H_
    int a = idx >> 10;          // / H_
    int h = idx & (H_ - 1);
    WhT[h * A_ + a] = Wh[idx];
}

// ---------------------------------------------------------------------------
// A0[b,a] = sum_x x0[b,x] * W0[a,x]   (33 MFLOP, negligible)
__global__ __launch_bounds__(256) void k_a0(const float* __restrict__ x0,
                                            const float* __restrict__ W0,
                                            float* __restrict__ A0) {
    int b = blockIdx.x >> 1;
    int a = ((blockIdx.x & 1) << 8) + threadIdx.x;
    const float* xr = x0 + b * X0_;
    const float* wr = W0 + a * X0_;
    float s = 0.f;
    #pragma unroll 4
    for (int x = 0; x < X0_; ++x) s += xr[x] * wr[x];
    A0[b * A_ + a] = s;
}

// ---------------------------------------------------------------------------
// Fused: scores[b,s] = relu(hidden[b,s,:]@Wh^T + bh + A0[b,:]) . V
// One WG = (batch b, 64 s-rows). 8 waves, each owns 64 N-columns:
// 4 M-tiles x 4 N-tiles of v_wmma_f32_16x16x4_f32 accumulators.
// Hidden tiles double-buffered in LDS via the Tensor Data Mover.
__global__ __launch_bounds__(256) void k_scores(const float* __restrict__ hidden,
                                                const float* __restrict__ WhT,
                                                const float* __restrict__ bh,
                                                const float* __restrict__ A0,
                                                const float* __restrict__ V,
                                                float* __restrict__ scores) {
    __shared__ float lds_h[2][ROWS * LDS_STRIDE];
    __shared__ float lds_s[ROWS];

    const int tid  = threadIdx.x;
    const int wave = tid >> 5;
    const int lane = tid & 31;
    const int m    = lane & 15;          // row-within-tile (A) / col-within-tile (B)
    const int kh   = (lane >> 4) << 1;   // lanes 16-31 hold K+2,K+3
    const int half = lane >> 4;

    const int b  = blockIdx.x >> 5;      // 32 s-tiles per batch
    const int s0 = (blockIdx.x & 31) * ROWS;
    const int n0 = wave * 64;            // this wave's first A-column

    const float* hbase = hidden + ((size_t)b * S_ + s0) * H_;

    if (tid < ROWS) lds_s[tid] = 0.f;

    v8f acc[4][4];
    #pragma unroll
    for (int mt = 0; mt < 4; ++mt)
        #pragma unroll
        for (int nt = 0; nt < 4; ++nt) acc[mt][nt] = {};

    // prologue: DMA chunk 0 into buffer 0 (wave 0 issues, TENSORcnt tracks)
    int cur = 0;
    if (tid < 32) {
        tdm_load_tile(hbase, (unsigned)(uintptr_t)(void*)&lds_h[0][0]);
        __builtin_amdgcn_s_wait_tensorcnt(0);
    }
    __syncthreads();

    #pragma unroll 1
    for (int kc = 0; kc < H_; kc += KC) {
        const int nxt = cur ^ 1;
        const bool have_next = (kc + KC) < H_;
        // overlap: DMA next chunk while all 8 waves run WMMA on current chunk
        if (tid < 32 && have_next)
            tdm_load_tile(hbase + (kc + KC),
                          (unsigned)(uintptr_t)(void*)&lds_h[nxt][0]);

        const float* lh = &lds_h[cur][0];
        #pragma unroll 2
        for (int k = 0; k < KC; k += 4) {
            // A fragments (shared rows, from LDS): lane m -> row, pair of K values
            v2f af[4];
            #pragma unroll
            for (int mt = 0; mt < 4; ++mt)
                af[mt] = *(const v2f*)&lh[((mt << 4) + m) * LDS_STRIDE + k + kh];

            // B fragments (WhT in L2): lane m -> column n0+nt*16+m, coalesced
            const int kg = kc + k + kh;
            v2f bf[4];
            #pragma unroll
            for (int nt = 0; nt < 4; ++nt) {
                int col = n0 + (nt << 4) + m;
                bf[nt].x = WhT[kg * A_ + col];
                bf[nt].y = WhT[(kg + 1) * A_ + col];
            }

            #pragma unroll
            for (int mt = 0; mt < 4; ++mt)
                #pragma unroll
                for (int nt = 0; nt < 4; ++nt)
                    acc[mt][nt] = __builtin_amdgcn_wmma_f32_16x16x4_f32(
                        false, af[mt], false, bf[nt],
                        (short)0, acc[mt][nt], false, false);
        }

        if (tid < 32 && have_next)
            __builtin_amdgcn_s_wait_tensorcnt(0);   // next chunk landed
        __syncthreads();                            // everyone done with cur
        cur = nxt;
    }

    // epilogue: + bh + A0, relu, * V, reduce over the A dimension
    float bias[4], vv[4];
    #pragma unroll
    for (int nt = 0; nt < 4; ++nt) {
        int col  = n0 + (nt << 4) + m;
        bias[nt] = bh[col] + A0[b * A_ + col];
        vv[nt]   = V[col];
    }

    #pragma unroll
    for (int mt = 0; mt < 4; ++mt) {
        #pragma unroll
        for (int r = 0; r < 8; ++r) {
            // VGPR r: lanes 0-15 -> row r, lanes 16-31 -> row 8+r (C/D layout)
            float v = 0.f;
            #pragma unroll
            for (int nt = 0; nt < 4; ++nt) {
                float x = acc[mt][nt][r] + bias[nt];
                x = x > 0.f ? x : 0.f;
                v += x * vv[nt];
            }
            atomicAdd(&lds_s[(mt << 4) + (half << 3) + r], v);
        }
    }
    __syncthreads();
    if (tid < ROWS) scores[(size_t)b * S_ + s0 + tid] = lds_s[tid];
}

// ---------------------------------------------------------------------------
// masked softmax over S per batch
__global__ __launch_bounds__(256) void k_softmax(const float* __restrict__ scores,
                                                 const int* __restrict__ lengths,
                                                 float* __restrict__ probs) {
    __shared__ float red[256];
    const int b   = blockIdx.x;
    const int tid = threadIdx.x;
    const int len = lengths[b];
    const float* sc = scores + (size_t)b * S_;

    const float NEG = -3.402823466e38f;
    float loc[8], mx = NEG;
    #pragma unroll
    for (int i = 0; i < 8; ++i) {
        int s = tid + (i << 8);
        float x = sc[s];
        loc[i] = (s < len) ? x : NEG;
        mx = fmaxf(mx, loc[i]);
    }
    red[tid] = mx; __syncthreads();
    for (int off = 128; off > 0; off >>= 1) {
        if (tid < off) red[tid] = fmaxf(red[tid], red[tid + off]);
        __syncthreads();
    }
    float gmx = red[0]; __syncthreads();

    float e[8], sum = 0.f;
    #pragma unroll
    for (int i = 0; i < 8; ++i) {
        int s = tid + (i << 8);
        e[i] = (s < len) ? __expf(loc[i] - gmx) : 0.f;
        sum += e[i];
    }
    red[tid] = sum; __syncthreads();
    for (int off = 128; off > 0; off >>= 1) {
        if (tid < off) red[tid] += red[tid + off];
        __syncthreads();
    }
    float inv = 1.f / red[0];
    #pragma unroll
    for (int i = 0; i < 8; ++i)
        probs[(size_t)b * S_ + tid + (i << 8)] = e[i] * inv;
}

// ---------------------------------------------------------------------------
// output[b,h] = sum_s probs[b,s] * hidden[b,s,h]  (streams hidden once, float4)
__global__ __launch_bounds__(256) void k_output(const float* __restrict__ hidden,
                                                const float* __restrict__ probs,
                                                float* __restrict__ out) {
    __shared__ float lp[S_];
    const int b   = blockIdx.x;
    const int tid = threadIdx.x;
    #pragma unroll
    for (int i = 0; i < 8; ++i)
        lp[tid + (i << 8)] = probs[(size_t)b * S_ + tid + (i << 8)];
    __syncthreads();

    const float* hb = hidden + (size_t)b * S_ * H_ + (tid << 2);
    float4 acc = make_float4(0.f, 0.f, 0.f, 0.f);
    #pragma unroll 4
    for (int s = 0; s < S_; ++s) {
        float p = lp[s];
        float4 hv = *(const float4*)(hb + (size_t)s * H_);
        acc.x += p * hv.x; acc.y += p * hv.y;
        acc.z += p * hv.z; acc.w += p * hv.w;
    }
    *(float4*)(out + b * H_ + (tid << 2)) = acc;
}

// ---------------------------------------------------------------------------
extern "C" void kernel_launch(void* const* d_in, const int* in_sizes, int n_in,
                              void* d_out, int out_size, void* d_ws, size_t ws_size,
                              hipStream_t stream) {
    const float* hidden  = (const float*)d_in[0];   // [B,S,H]
    const float* x0      = (const float*)d_in[1];   // [B,X0]
    const int*   lengths = (const int*)d_in[2];     // [B]
    const float* Wh      = (const float*)d_in[3];   // [A,H]
    const float* bh      = (const float*)d_in[4];   // [A]
    const float* W0      = (const float*)d_in[5];   // [A,X0]
    const float* V       = (const float*)d_in[6];   // [A]

    float* out   = (float*)d_out;        // [B,H] first
    float* probs = out + B_ * H_;        // then [B,S]

    float* ws     = (float*)d_ws;
    float* WhT    = ws + WS_WHT;
    float* A0     = ws + WS_A0;
    float* scores = ws + WS_SC;

    k_transpose<<<(A_ * H_) / 256, 256, 0, stream>>>(Wh, WhT);
    k_a0<<<B_ * (A_ / 256), 256, 0, stream>>>(x0, W0, A0);
    k_scores<<<B_ * (S_ / ROWS), 256, 0, stream>>>(hidden, WhT, bh, A0, V, scores);
    k_softmax<<<B_, 256, 0, stream>>>(scores, lengths, probs);
    k_output<<<B_, 256, 0, stream>>>(hidden, probs, out);
}